// MonoGAT_1855425872578
// MI455X (gfx1250) — compile-verified
//
#include <hip/hip_runtime.h>
#include <hip/hip_bf16.h>
#include <cmath>

// ---------------------------------------------------------------------------
// MonoGAT for MI455X (gfx1250, wave32).
//  - Dense h = x@W via V_WMMA_F32_16X16X4_F32, shapes baked in as template
//    params so every load is base+immediate and the K loop fully unrolls into
//    a clean 32x wmma stream per 16x16 tile.
//  - Edge softmax: pass1 atomic segment-max, pass2 unnormalized exp scatter
//    (p*h[src] -> acc[dst], p -> denom[dst]), pass3 divide + bias + act.
//    This phase is L2-atomic bound; features (25.6MB) are L2-resident.
// ---------------------------------------------------------------------------

typedef __attribute__((ext_vector_type(2))) float v2f;
typedef __attribute__((ext_vector_type(8))) float v8f;

#define N_NODES 50000
#define N_EDGES 800000
#define F_IN    128
#define HC      128      // HEADS*HID
#define HEADS   4
#define HID     32
#define N_CLS   16
#define NEG_SLOPE 0.2f

__device__ __forceinline__ void atomicMaxF32(float* addr, float val) {
  // Sign-split trick: totally ordered float max without CAS loops.
  if (val >= 0.0f) atomicMax((int*)addr, __float_as_int(val));
  else             atomicMin((unsigned int*)addr, __float_as_uint(val));
}

__device__ __forceinline__ float leaky(float v) {
  return v > 0.0f ? v : NEG_SLOPE * v;
}

// ---------------------------------------------------------------------------
// D[M,NCOLS] = A[M,K] * B[K,NCOLS] (row-major fp32). One wave -> one 16x16
// tile. Fragment layouts per ISA 7.12.2:
//   A 16x4:  lanes 0-15 row=lane, elems {K+0,K+1}; lanes 16-31 {K+2,K+3}.
//   B 4x16:  VGPR0 = rows {K+0 | K+2}, VGPR1 = rows {K+1 | K+3}, col=lane&15.
//   C/D:     VGPR r -> M = r + 8*(lane>=16), N = lane&15.
// ---------------------------------------------------------------------------
template <int K, int NCOLS>
__global__ void gemm_wmma_f32(const float* __restrict__ A,
                              const float* __restrict__ B,
                              float* __restrict__ D) {
  const int lane = threadIdx.x;       // 0..31
  const int half = lane >> 4;         // 0 or 1
  const int lid  = lane & 15;
  const int row0 = blockIdx.y * 16;
  const int col0 = blockIdx.x * 16;

  const float* arow = A + (size_t)(row0 + lid) * K + half * 2;   // 8B aligned
  const float* bcol = B + (size_t)(half * 2) * NCOLS + col0 + lid;

  v8f c = {0.f, 0.f, 0.f, 0.f, 0.f, 0.f, 0.f, 0.f};
#pragma unroll
  for (int k = 0; k < K; k += 4) {
    v2f a = *(const v2f*)(arow + k);          // global_load_b64, imm offset
    v2f b;
    b.x = bcol[k * NCOLS];                    // global_load_b32, imm offset
    b.y = bcol[(k + 1) * NCOLS];
    c = __builtin_amdgcn_wmma_f32_16x16x4_f32(false, a, false, b,
                                              (short)0, c, false, false);
  }
  float* drow = D + (size_t)(row0 + half * 8) * NCOLS + col0 + lid;
#pragma unroll
  for (int r = 0; r < 8; ++r) drow[r * NCOLS] = c[r];
}

// alpha_src[n,h] = sum_c h[n,h,c]*a_s[h,c] ; same for alpha_dst.
template <int H, int C>
__global__ void alpha_kernel(const float* __restrict__ feat,
                             const float* __restrict__ a_s,
                             const float* __restrict__ a_d,
                             float* __restrict__ as_out,
                             float* __restrict__ ad_out) {
  int n = blockIdx.x * blockDim.x + threadIdx.x;
  if (n >= N_NODES) return;
  const float* f = feat + (size_t)n * H * C;
#pragma unroll
  for (int h = 0; h < H; ++h) {
    float ss = 0.f, sd = 0.f;
#pragma unroll
    for (int c = 0; c < C; ++c) {
      float v = f[h * C + c];
      ss += v * a_s[h * C + c];
      sd += v * a_d[h * C + c];
    }
    as_out[n * H + h] = ss;
    ad_out[n * H + h] = sd;
  }
}

__global__ void fill_f32(float* __restrict__ p, float v, int n) {
  int t = blockIdx.x * blockDim.x + threadIdx.x;
  if (t < n) p[t] = v;
}

// Pass 1: segment max of leaky(alpha_s[src]+alpha_d[dst]) over dst.
// One thread per (edge-with-self-loops, head); H compile-time (no int div).
template <int H>
__global__ void edge_max_kernel(const long long* __restrict__ ei,
                                const float* __restrict__ as,
                                const float* __restrict__ ad,
                                float* __restrict__ emax) {
  int t = blockIdx.x * blockDim.x + threadIdx.x;
  if (t >= (N_EDGES + N_NODES) * H) return;
  int e = t / H, h = t - e * H;
  int s, d;
  if (e < N_EDGES) { s = (int)ei[e]; d = (int)ei[N_EDGES + e]; }
  else             { s = d = e - N_EDGES; }
  float v = leaky(as[s * H + h] + ad[d * H + h]);
  atomicMaxF32(&emax[(size_t)d * H + h], v);
}

// Pass 2 (HC=128, H=4, C=32): one block = one edge, thread = channel.
// Scatter p*h[src,c] into acc[dst,c]; one denom atomic per head.
__global__ void edge_agg128(const long long* __restrict__ ei,
                            const float* __restrict__ feat,
                            const float* __restrict__ as,
                            const float* __restrict__ ad,
                            const float* __restrict__ emax,
                            float* __restrict__ denom,
                            float* __restrict__ acc) {
  int e = blockIdx.x;
  int c = threadIdx.x;           // 0..127
  int h = c >> 5;
  int s, d;
  if (e < N_EDGES) { s = (int)ei[e]; d = (int)ei[N_EDGES + e]; }
  else             { s = d = e - N_EDGES; }
  float v = leaky(as[s * 4 + h] + ad[d * 4 + h]);
  float p = expf(v - emax[d * 4 + h]);
  atomicAdd(&acc[(size_t)d * HC + c], p * feat[(size_t)s * HC + c]);
  if ((c & 31) == 0) atomicAdd(&denom[d * 4 + h], p);
}

// Pass 2 for layer 3 (H=1, C=16): one thread per edge.
__global__ void edge_agg16(const long long* __restrict__ ei,
                           const float* __restrict__ feat,
                           const float* __restrict__ as,
                           const float* __restrict__ ad,
                           const float* __restrict__ emax,
                           float* __restrict__ denom,
                           float* __restrict__ acc) {
  int e = blockIdx.x * blockDim.x + threadIdx.x;
  if (e >= N_EDGES + N_NODES) return;
  int s, d;
  if (e < N_EDGES) { s = (int)ei[e]; d = (int)ei[N_EDGES + e]; }
  else             { s = d = e - N_EDGES; }
  float p = expf(leaky(as[s] + ad[d]) - emax[d]);
  const float* fs = feat + (size_t)s * N_CLS;
  float* ao = acc + (size_t)d * N_CLS;
#pragma unroll
  for (int c = 0; c < N_CLS; ++c) atomicAdd(&ao[c], p * fs[c]);
  atomicAdd(&denom[d], p);
}

// out = elu(acc/denom + bias), HC=128 layout.
__global__ void finalize_elu128(const float* __restrict__ acc,
                                const float* __restrict__ denom,
                                const float* __restrict__ bias,
                                float* __restrict__ out) {
  int t = blockIdx.x * blockDim.x + threadIdx.x;
  if (t >= N_NODES * HC) return;
  int n = t >> 7, c = t & 127, h = c >> 5;
  float v = acc[t] / (denom[n * 4 + h] + 1e-16f) + bias[c];
  out[t] = v > 0.f ? v : (expf(v) - 1.f);
}

// Final layer: divide, bias, row softmax over 16 classes.
__global__ void finalize_softmax16(const float* __restrict__ acc,
                                   const float* __restrict__ denom,
                                   const float* __restrict__ bias,
                                   float* __restrict__ out) {
  int n = blockIdx.x * blockDim.x + threadIdx.x;
  if (n >= N_NODES) return;
  float dv = denom[n] + 1e-16f;
  float v[N_CLS], m = -INFINITY;
#pragma unroll
  for (int c = 0; c < N_CLS; ++c) {
    v[c] = acc[(size_t)n * N_CLS + c] / dv + bias[c];
    m = fmaxf(m, v[c]);
  }
  float s = 0.f;
#pragma unroll
  for (int c = 0; c < N_CLS; ++c) { v[c] = expf(v[c] - m); s += v[c]; }
  float inv = 1.f / s;
#pragma unroll
  for (int c = 0; c < N_CLS; ++c) out[(size_t)n * N_CLS + c] = v[c] * inv;
}

extern "C" void kernel_launch(void* const* d_in, const int* in_sizes, int n_in,
                              void* d_out, int out_size, void* d_ws, size_t ws_size,
                              hipStream_t stream) {
  (void)in_sizes; (void)n_in; (void)out_size; (void)ws_size;
  const float*     x   = (const float*)d_in[0];
  const long long* ei  = (const long long*)d_in[1];   // int64 edge_index [2,E]
  const float* W1  = (const float*)d_in[2];
  const float* a1s = (const float*)d_in[3];
  const float* a1d = (const float*)d_in[4];
  const float* b1  = (const float*)d_in[5];
  const float* W2  = (const float*)d_in[6];
  const float* a2s = (const float*)d_in[7];
  const float* a2d = (const float*)d_in[8];
  const float* b2  = (const float*)d_in[9];
  const float* W3  = (const float*)d_in[10];
  const float* a3s = (const float*)d_in[11];
  const float* a3d = (const float*)d_in[12];
  const float* b3  = (const float*)d_in[13];

  // Workspace carve (floats): feat | acc | xnext | as | ad | emax | denom
  float* feat  = (float*)d_ws;
  float* acc   = feat  + (size_t)N_NODES * HC;
  float* xnext = acc   + (size_t)N_NODES * HC;
  float* asb   = xnext + (size_t)N_NODES * HC;
  float* adb   = asb   + (size_t)N_NODES * HEADS;
  float* emax  = adb   + (size_t)N_NODES * HEADS;
  float* denom = emax  + (size_t)N_NODES * HEADS;

  const int NE = N_EDGES + N_NODES;     // edges + self-loops
  const int THR = 256;

  // ---------------- Layer 1 ----------------
  gemm_wmma_f32<F_IN, HC><<<dim3(HC / 16, N_NODES / 16), 32, 0, stream>>>(x, W1, feat);
  alpha_kernel<HEADS, HID><<<(N_NODES + THR - 1) / THR, THR, 0, stream>>>(feat, a1s, a1d, asb, adb);
  fill_f32<<<(N_NODES * HEADS + THR - 1) / THR, THR, 0, stream>>>(emax, -INFINITY, N_NODES * HEADS);
  fill_f32<<<(N_NODES * HEADS + THR - 1) / THR, THR, 0, stream>>>(denom, 0.f, N_NODES * HEADS);
  fill_f32<<<(N_NODES * HC + THR - 1) / THR, THR, 0, stream>>>(acc, 0.f, N_NODES * HC);
  edge_max_kernel<HEADS><<<(NE * HEADS + THR - 1) / THR, THR, 0, stream>>>(ei, asb, adb, emax);
  edge_agg128<<<NE, HC, 0, stream>>>(ei, feat, asb, adb, emax, denom, acc);
  finalize_elu128<<<(N_NODES * HC + THR - 1) / THR, THR, 0, stream>>>(acc, denom, b1, xnext);

  // ---------------- Layer 2 ----------------
  gemm_wmma_f32<HC, HC><<<dim3(HC / 16, N_NODES / 16), 32, 0, stream>>>(xnext, W2, feat);
  alpha_kernel<HEADS, HID><<<(N_NODES + THR - 1) / THR, THR, 0, stream>>>(feat, a2s, a2d, asb, adb);
  fill_f32<<<(N_NODES * HEADS + THR - 1) / THR, THR, 0, stream>>>(emax, -INFINITY, N_NODES * HEADS);
  fill_f32<<<(N_NODES * HEADS + THR - 1) / THR, THR, 0, stream>>>(denom, 0.f, N_NODES * HEADS);
  fill_f32<<<(N_NODES * HC + THR - 1) / THR, THR, 0, stream>>>(acc, 0.f, N_NODES * HC);
  edge_max_kernel<HEADS><<<(NE * HEADS + THR - 1) / THR, THR, 0, stream>>>(ei, asb, adb, emax);
  edge_agg128<<<NE, HC, 0, stream>>>(ei, feat, asb, adb, emax, denom, acc);
  finalize_elu128<<<(N_NODES * HC + THR - 1) / THR, THR, 0, stream>>>(acc, denom, b2, xnext);

  // ---------------- Layer 3 (H=1, C=16) ----------------
  gemm_wmma_f32<HC, N_CLS><<<dim3(N_CLS / 16, N_NODES / 16), 32, 0, stream>>>(xnext, W3, feat);
  alpha_kernel<1, N_CLS><<<(N_NODES + THR - 1) / THR, THR, 0, stream>>>(feat, a3s, a3d, asb, adb);
  fill_f32<<<(N_NODES + THR - 1) / THR, THR, 0, stream>>>(emax, -INFINITY, N_NODES);
  fill_f32<<<(N_NODES + THR - 1) / THR, THR, 0, stream>>>(denom, 0.f, N_NODES);
  fill_f32<<<(N_NODES * N_CLS + THR - 1) / THR, THR, 0, stream>>>(acc, 0.f, N_NODES * N_CLS);
  edge_max_kernel<1><<<(NE + THR - 1) / THR, THR, 0, stream>>>(ei, asb, adb, emax);
  edge_agg16<<<(NE + THR - 1) / THR, THR, 0, stream>>>(ei, feat, asb, adb, emax, denom, acc);
  finalize_softmax16<<<(N_NODES + THR - 1) / THR, THR, 0, stream>>>(acc, denom, b3, (float*)d_out);
}